// NodeModel_17910013624369
// MI455X (gfx1250) — compile-verified
//
#include <hip/hip_runtime.h>

typedef __attribute__((ext_vector_type(2))) float v2f;
typedef __attribute__((ext_vector_type(8))) float v8f;

#define DD  128   // feature dim
#define DIN 259   // 2*D + F

// ---------------------------------------------------------------------------
// Kernel 1: zero the aggregation buffer (workspace is poisoned, not zeroed)
// ---------------------------------------------------------------------------
__global__ void zero_f32(float* __restrict__ p, int n) {
  int i = blockIdx.x * 256 + threadIdx.x;
  if (i < n) p[i] = 0.0f;
}

// ---------------------------------------------------------------------------
// Kernel 2: scatter-add  agg[dest[e], :] += edge_attr[e, :]
// One wave per edge: lane c reads a float4 chunk (fully coalesced 512B/wave),
// then 4 global_atomic_add_f32 (unsafeAtomicAdd -> native FP32 atomics).
// This phase is the HBM roofline term (~820 MB of edge_attr at 23.3 TB/s).
// ---------------------------------------------------------------------------
__global__ void scatter_add(const float* __restrict__ ea,
                            const int*   __restrict__ ei,
                            float*       __restrict__ agg, int E) {
  int idx = blockIdx.x * 256 + threadIdx.x;
  int e = idx >> 5;
  if (e >= E) return;
  int c = (idx & 31) * 4;
  int dst = ei[E + e];                       // edge_index row 1 = dest
  const float4 v = *(const float4*)(ea + (size_t)e * DD + c);
  float* base = agg + (size_t)dst * DD + c;
  unsafeAtomicAdd(base + 0, v.x);
  unsafeAtomicAdd(base + 1, v.y);
  unsafeAtomicAdd(base + 2, v.z);
  unsafeAtomicAdd(base + 3, v.w);
}

__device__ __forceinline__ float silu(float v) {
  return v / (1.0f + __expf(-v));
}

__device__ __forceinline__ v8f wmma_f32(v2f a, v2f b, v8f c) {
  return __builtin_amdgcn_wmma_f32_16x16x4_f32(false, a, false, b,
                                               (short)0, c, false, false);
}

// ---------------------------------------------------------------------------
// Kernel 3: fused 3-layer MLP with V_WMMA_F32_16X16X4_F32.
// 5 waves / block; each wave owns TWO 16-row tiles (M=32) and all N=128
// outputs (16 accumulator tiles = 128 VGPRs), so every B fragment load feeds
// two WMMAs. 100000 rows = 625 blocks * 5 waves * 32 rows exactly: no guards,
// EXEC statically all-ones (WMMA requirement). The layer-0 K loop is split
// into branch-free x / agg regions plus one select-based tail step for f.
// Inter-layer D->A transpose goes through a wave-private 32x128 LDS buffer.
// ---------------------------------------------------------------------------
__global__ __launch_bounds__(160) void fused_mlp(
    const float* __restrict__ x,  const float* __restrict__ agg,
    const float* __restrict__ f,
    const float* __restrict__ W0, const float* __restrict__ b0,
    const float* __restrict__ W1, const float* __restrict__ b1,
    const float* __restrict__ W2, const float* __restrict__ b2,
    float* __restrict__ out)
{
  __shared__ float lds[5 * 32 * DD];           // 80 KB of the 320 KB WGP LDS
  const int lane = threadIdx.x & 31;
  const int wave = threadIdx.x >> 5;
  const int half = lane >> 4;                  // K/lane-half selector
  const int mn   = lane & 15;                  // row (A) / col (B,D) index
  const int tile = blockIdx.x * 5 + wave;      // 32-row tile id (exact grid)
  float* hbuf = lds + wave * (32 * DD);

  const int row0 = tile * 32 + mn;             // A rows for sub-tile 0
  const int row1 = row0 + 16;                  // A rows for sub-tile 1
  const int kofs = 2 * half;                   // lane's K pair base

  v8f acc0[8], acc1[8];

  // ==================== Layer 0: h = silu([x|agg|f] @ W0 + b0) ============
  #pragma unroll
  for (int nt = 0; nt < 8; ++nt) {
    float bv = b0[nt * 16 + mn];
    v8f t;
    #pragma unroll
    for (int r = 0; r < 8; ++r) t[r] = bv;
    acc0[nt] = t;
    acc1[nt] = t;
  }

  // --- region 1: k in [0,128) from x (branch-free) ---
  {
    const float* ap0 = x + (size_t)row0 * DD + kofs;
    const float* ap1 = x + (size_t)row1 * DD + kofs;
    const float* wp  = W0 + (size_t)kofs * DD + mn;
    #pragma unroll 2
    for (int k0 = 0; k0 < 128; k0 += 4) {
      v2f a0 = *(const v2f*)(ap0 + k0);
      v2f a1 = *(const v2f*)(ap1 + k0);
      const float* wr = wp + (size_t)k0 * DD;
      #pragma unroll
      for (int nt = 0; nt < 8; ++nt) {
        v2f b = { wr[nt * 16], wr[DD + nt * 16] };
        acc0[nt] = wmma_f32(a0, b, acc0[nt]);
        acc1[nt] = wmma_f32(a1, b, acc1[nt]);
      }
    }
  }

  // --- region 2: k in [128,256) from agg (branch-free) ---
  {
    const float* ap0 = agg + (size_t)row0 * DD + kofs;
    const float* ap1 = agg + (size_t)row1 * DD + kofs;
    const float* wp  = W0 + (size_t)(128 + kofs) * DD + mn;
    #pragma unroll 2
    for (int k0 = 0; k0 < 128; k0 += 4) {
      v2f a0 = *(const v2f*)(ap0 + k0);
      v2f a1 = *(const v2f*)(ap1 + k0);
      const float* wr = wp + (size_t)k0 * DD;
      #pragma unroll
      for (int nt = 0; nt < 8; ++nt) {
        v2f b = { wr[nt * 16], wr[DD + nt * 16] };
        acc0[nt] = wmma_f32(a0, b, acc0[nt]);
        acc1[nt] = wmma_f32(a1, b, acc1[nt]);
      }
    }
  }

  // --- tail: k in [256,260), f has 3 cols, k=259 padded with zeros ---
  {
    const float* fr0 = f + row0 * 3;
    const float* fr1 = f + row1 * 3;
    v2f a0, a1;
    a0.x = half ? fr0[2] : fr0[0];
    a0.y = half ? 0.0f   : fr0[1];
    a1.x = half ? fr1[2] : fr1[0];
    a1.y = half ? 0.0f   : fr1[1];
    const float* wr = W0 + (size_t)(256 + kofs) * DD + mn;  // rows 256/258
    #pragma unroll
    for (int nt = 0; nt < 8; ++nt) {
      v2f b;
      b.x = wr[nt * 16];                               // k=256 or 258 (valid)
      b.y = half ? 0.0f : wr[DD + nt * 16];            // k=257 valid, 259 pad
      acc0[nt] = wmma_f32(a0, b, acc0[nt]);
      acc1[nt] = wmma_f32(a1, b, acc1[nt]);
    }
  }

  // silu -> LDS (D layout: row = 8*half + r, col = mn)
  #pragma unroll
  for (int nt = 0; nt < 8; ++nt) {
    #pragma unroll
    for (int r = 0; r < 8; ++r) {
      hbuf[(half * 8 + r) * DD + nt * 16 + mn]        = silu(acc0[nt][r]);
      hbuf[(16 + half * 8 + r) * DD + nt * 16 + mn]   = silu(acc1[nt][r]);
    }
  }

  // ==================== Layer 1: h = silu(h @ W1 + b1) ====================
  #pragma unroll
  for (int nt = 0; nt < 8; ++nt) {
    float bv = b1[nt * 16 + mn];
    v8f t;
    #pragma unroll
    for (int r = 0; r < 8; ++r) t[r] = bv;
    acc0[nt] = t;
    acc1[nt] = t;
  }
  {
    const float* ap0 = hbuf + mn * DD + kofs;
    const float* ap1 = hbuf + (16 + mn) * DD + kofs;
    const float* wp  = W1 + (size_t)kofs * DD + mn;
    #pragma unroll 2
    for (int k0 = 0; k0 < 128; k0 += 4) {
      v2f a0 = *(const v2f*)(ap0 + k0);                // ds_load_b64
      v2f a1 = *(const v2f*)(ap1 + k0);
      const float* wr = wp + (size_t)k0 * DD;
      #pragma unroll
      for (int nt = 0; nt < 8; ++nt) {
        v2f b = { wr[nt * 16], wr[DD + nt * 16] };
        acc0[nt] = wmma_f32(a0, b, acc0[nt]);
        acc1[nt] = wmma_f32(a1, b, acc1[nt]);
      }
    }
  }
  #pragma unroll
  for (int nt = 0; nt < 8; ++nt) {
    #pragma unroll
    for (int r = 0; r < 8; ++r) {
      hbuf[(half * 8 + r) * DD + nt * 16 + mn]        = silu(acc0[nt][r]);
      hbuf[(16 + half * 8 + r) * DD + nt * 16 + mn]   = silu(acc1[nt][r]);
    }
  }

  // ==================== Layer 2: out = h @ W2 + b2 ========================
  #pragma unroll
  for (int nt = 0; nt < 8; ++nt) {
    float bv = b2[nt * 16 + mn];
    v8f t;
    #pragma unroll
    for (int r = 0; r < 8; ++r) t[r] = bv;
    acc0[nt] = t;
    acc1[nt] = t;
  }
  {
    const float* ap0 = hbuf + mn * DD + kofs;
    const float* ap1 = hbuf + (16 + mn) * DD + kofs;
    const float* wp  = W2 + (size_t)kofs * DD + mn;
    #pragma unroll 2
    for (int k0 = 0; k0 < 128; k0 += 4) {
      v2f a0 = *(const v2f*)(ap0 + k0);
      v2f a1 = *(const v2f*)(ap1 + k0);
      const float* wr = wp + (size_t)k0 * DD;
      #pragma unroll
      for (int nt = 0; nt < 8; ++nt) {
        v2f b = { wr[nt * 16], wr[DD + nt * 16] };
        acc0[nt] = wmma_f32(a0, b, acc0[nt]);
        acc1[nt] = wmma_f32(a1, b, acc1[nt]);
      }
    }
  }
  {
    float* op = out + (size_t)(tile * 32) * DD + mn;
    #pragma unroll
    for (int nt = 0; nt < 8; ++nt) {
      #pragma unroll
      for (int r = 0; r < 8; ++r) {
        op[(size_t)(half * 8 + r) * DD + nt * 16]      = acc0[nt][r];
        op[(size_t)(16 + half * 8 + r) * DD + nt * 16] = acc1[nt][r];
      }
    }
  }
}

// ---------------------------------------------------------------------------
// Host-side launch
// ---------------------------------------------------------------------------
extern "C" void kernel_launch(void* const* d_in, const int* in_sizes, int n_in,
                              void* d_out, int out_size, void* d_ws, size_t ws_size,
                              hipStream_t stream) {
  const float* x   = (const float*)d_in[0];
  const int*   ei  = (const int*)  d_in[1];
  const float* ea  = (const float*)d_in[2];
  const float* f   = (const float*)d_in[3];
  const float* W0  = (const float*)d_in[4];
  const float* b0  = (const float*)d_in[5];
  const float* W1  = (const float*)d_in[6];
  const float* b1  = (const float*)d_in[7];
  const float* W2  = (const float*)d_in[8];
  const float* b2  = (const float*)d_in[9];
  float* out = (float*)d_out;

  const int N = in_sizes[0] / DD;      // 100000
  const int E = in_sizes[1] / 2;       // 1600000
  float* agg = (float*)d_ws;           // N*DD*4 = 51.2 MB scratch

  const int n_agg = N * DD;
  zero_f32<<<(n_agg + 255) / 256, 256, 0, stream>>>(agg, n_agg);

  const long long sthreads = (long long)E * 32;
  scatter_add<<<(int)((sthreads + 255) / 256), 256, 0, stream>>>(ea, ei, agg, E);

  // 100000 rows = 625 blocks * 5 waves * 32 rows -- exact, no tail guard
  const int n_blocks = N / (32 * 5);
  fused_mlp<<<n_blocks, 160, 0, stream>>>(
      x, agg, f, W0, b0, W1, b1, W2, b2, out);
}